// GMLoss_25039659336371
// MI455X (gfx1250) — compile-verified
//
#include <hip/hip_runtime.h>

// CDNA5 / gfx1250: wave32, WMMA (not MFMA).
typedef float v2f __attribute__((ext_vector_type(2)));
typedef float v8f __attribute__((ext_vector_type(8)));

#define BATCH 8
#define DIMS  3
#define NPTS  4096
#define TILE  16
#define WAVES_PER_BLOCK 8

// For each batch b and each row point m of X: out[b*N + m] = min_n ||X[:,m] - Y[:,n]||^2.
// X, Y are [B, D, N] (d-major within batch), the reference layout.
// One wave32 per 16-row tile. The K=4 slot of V_WMMA_F32_16X16X4_F32 absorbs the
// distance expansion: A' = [-2*s0,-2*s1,-2*s2, 1], B' = [t0,t1,t2, t^2]
//   => WMMA(A',B') = -2*(s . t) + t^2 ;  P = that + s^2 (loop-invariant per lane).
__global__ __launch_bounds__(32 * WAVES_PER_BLOCK)
void chamfer_rowmin_kernel(const float* __restrict__ X,
                           const float* __restrict__ Y,
                           float* __restrict__ rowmin_out)
{
    const int lane  = threadIdx.x & 31;
    const int wave  = threadIdx.x >> 5;
    const int tile  = blockIdx.x * WAVES_PER_BLOCK + wave;   // global row-tile id
    const int tilesPerBatch = NPTS / TILE;                   // 256
    const int b  = tile / tilesPerBatch;
    const int m0 = (tile % tilesPerBatch) * TILE;

    const int j    = lane & 15;   // row (A) / col (B) index within tile
    const int half = lane >> 4;   // K half: k in {0,1} or {2,3}
    const int d0   = 2 * half;    // first real dim this half loads (0 or 2)
    const int d1   = half ? 2 : 1; // clamped second row: half1's .y is replaced anyway

    const float* __restrict__ Xb = X + (size_t)b * DIMS * NPTS;
    const float* __restrict__ Yb = Y + (size_t)b * DIMS * NPTS;
    const float* __restrict__ Y0 = Yb + (size_t)d0 * NPTS;   // unconditional row ptrs
    const float* __restrict__ Y1 = Yb + (size_t)d1 * NPTS;

    // ---- A fragment (raw loads, branchless pad)
    const float sx = Xb[d0 * NPTS + m0 + j];
    const float sy = Xb[d1 * NPTS + m0 + j];
    const float sym = half ? 0.0f : sy;                      // pad contribution -> 0

    // s_sq[m0+j]: combine the two K-halves across lane pairs (L, L^16)
    const float ssq_part = fmaf(sym, sym, sx * sx);
    const float ssq      = ssq_part + __shfl_xor(ssq_part, 16);
    // Redistribute to the C/D layout: lane L, vgpr v needs s_sq at m = v + 8*half
    float ssq_m[8];
#pragma unroll
    for (int v = 0; v < 8; ++v)
        ssq_m[v] = __shfl(ssq, v + 8 * half);

    // A' = [-2*s0, -2*s1, -2*s2, 1]  (k=3 slot = upper half's .y)
    v2f a;
    a.x = -2.0f * sx;
    a.y = half ? 1.0f : -2.0f * sy;

    float rmin[8];
#pragma unroll
    for (int v = 0; v < 8; ++v) rmin[v] = 3.402823466e38f;

    const v8f czero = {};

    for (int n0 = 0; n0 < NPTS; n0 += TILE) {
        // ---- B' fragment: unconditional loads, no exec divergence
        const float ty0 = Y0[n0 + j];
        const float ty1 = Y1[n0 + j];
        const float tym = half ? 0.0f : ty1;

        const float tpart = fmaf(tym, tym, ty0 * ty0);
        const float tsq   = tpart + __shfl_xor(tpart, 16);   // full t^2 for column j

        v2f bb;
        bb.x = ty0;
        bb.y = half ? tsq : ty1;                             // k=3 slot carries t^2

        // D = A' x B' (+0) = -2*cross + t^2, full f32
        // 8 args: (neg_a, A, neg_b, B, c_mod, C, reuse_a, reuse_b)
        v8f c = __builtin_amdgcn_wmma_f32_16x16x4_f32(
            false, a, false, bb, (short)0, czero, false, false);

#pragma unroll
        for (int v = 0; v < 8; ++v)
            rmin[v] = fminf(rmin[v], c[v] + ssq_m[v]);       // P = s^2 - 2*cross + t^2
    }

    // Min-reduce across the 16 lanes of each half (masks < 16 keep halves separate)
#pragma unroll
    for (int mask = 1; mask <= 8; mask <<= 1) {
#pragma unroll
        for (int v = 0; v < 8; ++v)
            rmin[v] = fminf(rmin[v], __shfl_xor(rmin[v], mask));
    }

    // Lane 0 writes rows m0..m0+7; lane 16 writes rows m0+8..m0+15.
    if (j == 0) {
        float* __restrict__ out = rowmin_out + (size_t)b * NPTS + m0 + 8 * half;
#pragma unroll
        for (int v = 0; v < 8; ++v) out[v] = rmin[v];
    }
}

// Geman-McClure (MU=1) + means:
// loss = sum over all 2*B*N mins of d/(d+1), divided by B*N.
__global__ __launch_bounds__(256)
void gm_loss_reduce_kernel(const float* __restrict__ dists,
                           float* __restrict__ out, int n)
{
    __shared__ float sdata[256];
    float acc = 0.0f;
    for (int i = threadIdx.x; i < n; i += 256) {
        float d = dists[i];
        acc += d / (d + 1.0f);   // MU * d / (d + MU), MU = 1
    }
    sdata[threadIdx.x] = acc;
    __syncthreads();
#pragma unroll
    for (int s = 128; s > 0; s >>= 1) {
        if (threadIdx.x < s) sdata[threadIdx.x] += sdata[threadIdx.x + s];
        __syncthreads();
    }
    if (threadIdx.x == 0) out[0] = sdata[0] * (1.0f / (float)(BATCH * NPTS));
}

extern "C" void kernel_launch(void* const* d_in, const int* in_sizes, int n_in,
                              void* d_out, int out_size, void* d_ws, size_t ws_size,
                              hipStream_t stream) {
    const float* srcs = (const float*)d_in[0];  // [B, D, N] f32
    const float* tgts = (const float*)d_in[1];  // [B, D, N] f32
    (void)in_sizes; (void)n_in; (void)out_size; (void)ws_size;

    float* d_src_min = (float*)d_ws;                 // [B*N] min over tgt (axis=2)
    float* d_tgt_min = d_src_min + BATCH * NPTS;     // [B*N] min over src (axis=1)

    const int rowTiles = BATCH * (NPTS / TILE);      // 2048 waves
    dim3 block(32 * WAVES_PER_BLOCK);
    dim3 grid(rowTiles / WAVES_PER_BLOCK);           // 256 blocks

    chamfer_rowmin_kernel<<<grid, block, 0, stream>>>(srcs, tgts, d_src_min);
    chamfer_rowmin_kernel<<<grid, block, 0, stream>>>(tgts, srcs, d_tgt_min);
    gm_loss_reduce_kernel<<<1, 256, 0, stream>>>(d_src_min, (float*)d_out,
                                                 2 * BATCH * NPTS);
}